// LayeredNANDGraph_24395414241301
// MI455X (gfx1250) — compile-verified
//
#include <hip/hip_runtime.h>
#include <hip/hip_bf16.h>

// ---------------- problem constants (fixed by the reference) ----------------
#define NUM_INPUTS 1024
#define NUM_OUT    1024
#define NN         2048   // neurons per hidden layer
#define L_HID      5
#define WW         2048   // int32 words per node (8 KB per row)
#define BB         8      // batch size

// ---------------- gfx1250 feature probes ----------------
#if defined(__has_builtin)
#if __has_builtin(__builtin_amdgcn_global_load_async_to_lds_b128)
#define HAVE_ASYNC_BUILTIN 1
#endif
#if __has_builtin(__builtin_amdgcn_s_wait_asynccnt)
#define HAVE_WAITASYNC_BUILTIN 1
#endif
#if __has_builtin(__builtin_amdgcn_tensor_load_to_lds) && __has_builtin(__builtin_amdgcn_s_wait_tensorcnt)
#define HAVE_TDM_BUILTIN 1
#endif
#endif

typedef unsigned int u32;
typedef __attribute__((ext_vector_type(4))) int v4i_t;
typedef __attribute__((ext_vector_type(4))) u32 v4u_t;
typedef __attribute__((ext_vector_type(8))) int v8i_t;
typedef __attribute__((address_space(1))) v4i_t gv4i_t;   // global int4
typedef __attribute__((address_space(3))) v4i_t lv4i_t;   // LDS int4

// ---- per-lane async copy engine (ASYNCcnt): global -> LDS, 16 B granules ----
__device__ __forceinline__ void async_g2l_b128(const int* gsrc, int* ldst) {
#ifdef HAVE_ASYNC_BUILTIN
  const v4i_t* ggen = (const v4i_t*)gsrc;
  v4i_t*       lgen = (v4i_t*)ldst;
  gv4i_t* gp = (gv4i_t*)ggen;   // flat -> global (drops const)
  lv4i_t* lp = (lv4i_t*)lgen;   // flat -> LDS
  __builtin_amdgcn_global_load_async_to_lds_b128(gp, lp, /*offset=*/0, /*cpol=*/0);
#else
  unsigned loff = (unsigned)(size_t)(__attribute__((address_space(3))) int*)ldst;
  asm volatile("global_load_async_to_lds_b128 %0, %1, off"
               :: "v"(loff), "v"(gsrc) : "memory");
#endif
}

__device__ __forceinline__ void wait_async0() {
#ifdef HAVE_WAITASYNC_BUILTIN
  __builtin_amdgcn_s_wait_asynccnt(0);
#else
  asm volatile("s_wait_asynccnt 0" ::: "memory");
#endif
}

#ifdef HAVE_TDM_BUILTIN
// ---- Tensor Data Mover (TENSORcnt): one DMA moves a whole 8 KB row to LDS ----
// D# per CDNA5 ISA ch.8: group0 = {count/flags, lds_addr, global_addr, type},
// group1 = {data_size, tensor dims/strides, tile dims}. 1-D tile of WW dwords.
// This toolchain exposes the 6-arg builtin form:
//   (uint32x4 g0, int32x8 g1, int32x4 g2, int32x4 g3, int32x8, i32 cpol)
__device__ __forceinline__ void tdm_load_row(const int* gsrc, unsigned lds_off) {
  const unsigned long long ga = (unsigned long long)(size_t)gsrc;
  v4u_t g0;
  g0.x = 1u;                                 // count=1 (valid user descriptor)
  g0.y = lds_off;                            // LDS byte address of tile start
  g0.z = (u32)ga;                            // global_addr[31:0]
  g0.w = ((u32)(ga >> 32) & 0x01FFFFFFu) |   // global_addr[56:32]
         (2u << 30);                         // type = 2 ("image")
  v8i_t g1;
  g1.s0 = (int)(2u << 16);                   // workgroup_mask=0, data_size=2 (4 B)
  g1.s1 = (int)((WW & 0xFFFFu) << 16);       // tensor_dim0[15:0]  (bits 79:48 = 2048)
  g1.s2 = (int)(1u << 16);                   // tensor_dim0[31:16]=0, tensor_dim1=1
  g1.s3 = (int)((WW & 0xFFFFu) << 16);       // tensor_dim1[31:16]=0, tile_dim0=2048
  g1.s4 = 0;                                 // tile_dim1=0 (unused), tile_dim2=0
  g1.s5 = (int)WW;                           // tensor_dim0_stride[31:0] = 2048
  g1.s6 = (int)((WW & 0xFFFFu) << 16);       // dim0_stride[47:32]=0, dim1_stride lo
  g1.s7 = 0;                                 // tensor_dim1_stride hi
  const v4i_t z4 = {0, 0, 0, 0};             // groups 2/3 unused (<=2-D tensor)
  const v8i_t z8 = {0, 0, 0, 0, 0, 0, 0, 0};
  __builtin_amdgcn_tensor_load_to_lds(g0, g1, z4, z4, z8, /*cpol=*/0);
}
#endif

// ---------------- stateless counter-based RNG (deterministic across replays) ----------------
__device__ __forceinline__ unsigned mix4(unsigned a, unsigned b, unsigned c, unsigned d) {
  unsigned h = a * 0x9E3779B1u;
  h ^= b + 0x85EBCA77u + (h << 6) + (h >> 2);
  h ^= c + 0xC2B2AE3Du + (h << 6) + (h >> 2);
  h ^= d + 0x27D4EB2Fu + (h << 6) + (h >> 2);
  h ^= h >> 16; h *= 0x7FEB352Du; h ^= h >> 15; h *= 0x846CA68Bu; h ^= h >> 16;
  return h;
}

// ============================================================================
// Kernel 1: categorical (Gumbel-max) + Bernoulli sampling for one layer.
// grid = (n_out, 2); block (o, k) computes argmax_i(logit[k][o][i] + G) for all
// B batch samples in a single pass over the logit row (row read once, B noises).
// wave32 shuffle reduction + LDS cross-wave reduction.
// ============================================================================
__global__ void __launch_bounds__(256)
sample_kernel(const float* __restrict__ adj,   // [2, n_out, n_in]
              const float* __restrict__ invl,  // [n_out]
              int* __restrict__ idxbuf,        // [2, BB, n_out]
              int* __restrict__ invbuf,        // [n_out]
              int n_in, int n_out, unsigned seed)
{
  const int o   = blockIdx.x;
  const int k   = blockIdx.y;
  const int tid = threadIdx.x;
  const float* row = adj + ((size_t)k * n_out + o) * n_in;

  float bv[BB]; int bi[BB];
#pragma unroll
  for (int b = 0; b < BB; ++b) { bv[b] = -3.4e38f; bi[b] = 0; }

  for (int i = tid; i < n_in; i += 256) {
    const float logit = row[i];
#pragma unroll
    for (int b = 0; b < BB; ++b) {
      const unsigned h = mix4(seed, (unsigned)((k << 16) | o), (unsigned)i, (unsigned)b);
      const float u = (float)(h >> 8) * 5.9604645e-08f + 2.9802322e-08f; // (0,1)
      const float g = -__logf(-__logf(u));                               // Gumbel(0,1)
      const float v = logit + g;
      if (v > bv[b]) { bv[b] = v; bi[b] = i; }
    }
  }

  // intra-wave reduction (wave32: warpSize == 32 on gfx1250)
#pragma unroll
  for (int off = 16; off > 0; off >>= 1) {
#pragma unroll
    for (int b = 0; b < BB; ++b) {
      const float ov = __shfl_xor(bv[b], off, 32);
      const int   oi = __shfl_xor(bi[b], off, 32);
      if (ov > bv[b]) { bv[b] = ov; bi[b] = oi; }
    }
  }

  __shared__ float swv[8][BB];
  __shared__ int   swi[8][BB];
  const int wave = tid >> 5;        // 256 threads = 8 wave32
  if ((tid & 31) == 0) {
#pragma unroll
    for (int b = 0; b < BB; ++b) { swv[wave][b] = bv[b]; swi[wave][b] = bi[b]; }
  }
  __syncthreads();
  if (tid < BB) {
    float best = swv[0][tid]; int bidx = swi[0][tid];
#pragma unroll
    for (int w = 1; w < 8; ++w)
      if (swv[w][tid] > best) { best = swv[w][tid]; bidx = swi[w][tid]; }
    idxbuf[((size_t)k * BB + tid) * n_out + o] = bidx;
  }
  if (k == 0 && tid == 0) {
    const unsigned h = mix4(seed ^ 0xA511E9B3u, (unsigned)o, 0x51ED270Bu, 0u);
    const float u = (float)(h >> 8) * 5.9604645e-08f;
    const float p = 1.0f / (1.0f + __expf(-invl[o]));   // sigmoid
    invbuf[o] = (u < p) ? 1 : 0;
  }
}

// ============================================================================
// Kernel 2: gather two selected 8 KB bit-rows, AND, optional complement.
// grid = (n_out, B); one workgroup per output row. The two (workgroup-uniform)
// rows are staged into LDS on the two independent gfx1250 DMA engines:
//   row0 -> Tensor Data Mover (one tensor_load_to_lds, TENSORcnt, wave 0 only)
//   row1 -> per-lane async copy (global_load_async_to_lds_b128, ASYNCcnt)
// then combined with ds_load_b128 -> v_and/v_xor -> global_store_b128.
// Purely bandwidth bound; one layer's activation (128 MB) is L2-resident.
// ============================================================================
__global__ void __launch_bounds__(256)
nand_gather_kernel(const int* __restrict__ prev,
                   const int* __restrict__ idxbuf,   // [2, BB, n_out]
                   const int* __restrict__ invbuf,   // [n_out]
                   int* __restrict__ out,            // [BB, n_out, WW]
                   int n_in, int n_out, int prev_batched)
{
  __shared__ int lds0[WW];   // 8 KB
  __shared__ int lds1[WW];   // 8 KB
  const int o   = blockIdx.x;
  const int b   = blockIdx.y;
  const int tid = threadIdx.x;

  // row indices are uniform across the block -> force scalar for TDM descriptor
  const int i0 = __builtin_amdgcn_readfirstlane(idxbuf[(0 * BB + b) * n_out + o]);
  const int i1 = __builtin_amdgcn_readfirstlane(idxbuf[(1 * BB + b) * n_out + o]);
  const unsigned m = invbuf[o] ? 0xFFFFFFFFu : 0u;

  const size_t r0base = prev_batched ? ((size_t)b * n_in + (size_t)i0) : (size_t)i0;
  const size_t r1base = prev_batched ? ((size_t)b * n_in + (size_t)i1) : (size_t)i1;
  const int* r0 = prev + r0base * WW;
  const int* r1 = prev + r1base * WW;

#ifdef HAVE_TDM_BUILTIN
  // row0: single TDM DMA for the whole 8 KB row. TDM ignores EXEC, so gate it
  // with a wave-uniform branch (only wave 0 reaches the instruction at all).
  if (tid < 32) {
    const unsigned lds_off0 =
        (unsigned)(size_t)(__attribute__((address_space(3))) int*)&lds0[0];
    tdm_load_row(r0, lds_off0);
    __builtin_amdgcn_s_wait_tensorcnt(0);
  }
#else
#pragma unroll
  for (int c = tid; c < WW / 4; c += 256) {
    async_g2l_b128(r0 + 4 * c, &lds0[4 * c]);
  }
#endif

  // row1: per-lane async copies, overlapping the TDM transfer on ASYNCcnt
#pragma unroll
  for (int c = tid; c < WW / 4; c += 256) {       // 512 int4 chunks, 2 per thread
    async_g2l_b128(r1 + 4 * c, &lds1[4 * c]);
  }
  wait_async0();       // this wave's async LDS writes are complete
  __syncthreads();     // all waves' portions (and the TDM tile) visible

  int* orow = out + ((size_t)b * n_out + o) * WW;
#pragma unroll
  for (int c = tid; c < WW / 4; c += 256) {
    const int4 a = *(const int4*)&lds0[4 * c];
    const int4 x = *(const int4*)&lds1[4 * c];
    int4 r;
    r.x = (int)(((unsigned)(a.x & x.x)) ^ m);
    r.y = (int)(((unsigned)(a.y & x.y)) ^ m);
    r.z = (int)(((unsigned)(a.z & x.z)) ^ m);
    r.w = (int)(((unsigned)(a.w & x.w)) ^ m);
    *(int4*)&orow[4 * c] = r;
  }
}

// ============================================================================
// Orchestration: 7 layers, each = sample + gather, stream-ordered.
// Workspace layout (needs ~256.2 MB): ping-pong activations + idx/invert.
// ============================================================================
extern "C" void kernel_launch(void* const* d_in, const int* in_sizes, int n_in_args,
                              void* d_out, int out_size, void* d_ws, size_t ws_size,
                              hipStream_t stream) {
  const int*   inputs = (const int*)d_in[0];   // [1024, 2048] int32
  const float* adj0   = (const float*)d_in[1]; // [2, 2048, 1024]
  const float* inv0   = (const float*)d_in[2]; // [2048]
  const float* adjh   = (const float*)d_in[3]; // [5, 2, 2048, 2048]
  const float* invh   = (const float*)d_in[4]; // [5, 2048]
  const float* adjl   = (const float*)d_in[5]; // [2, 1024, 2048]
  const float* invl   = (const float*)d_in[6]; // [1024]
  int* out = (int*)d_out;                      // [8, 1024, 2048] int32

  const size_t actElems = (size_t)BB * NN * WW;        // 33,554,432 ints = 128 MB
  int* bufA   = (int*)d_ws;
  int* bufB   = bufA + actElems;
  int* idxbuf = bufB + actElems;                       // 2*BB*NN ints
  int* invbuf = idxbuf + (size_t)2 * BB * NN;          // NN ints

  // layer 0: 1024 inputs -> 2048 gates (prev shared across batch)
  sample_kernel<<<dim3(NN, 2), 256, 0, stream>>>(adj0, inv0, idxbuf, invbuf,
                                                 NUM_INPUTS, NN, 0u);
  nand_gather_kernel<<<dim3(NN, BB), 256, 0, stream>>>(inputs, idxbuf, invbuf, bufA,
                                                       NUM_INPUTS, NN, 0);

  // 5 hidden layers: 2048 -> 2048 (batched activations, ping-pong)
  int* cur = bufA; int* nxt = bufB;
  for (int l = 0; l < L_HID; ++l) {
    sample_kernel<<<dim3(NN, 2), 256, 0, stream>>>(
        adjh + (size_t)l * 2 * NN * NN, invh + (size_t)l * NN,
        idxbuf, invbuf, NN, NN, (unsigned)(l + 1));
    nand_gather_kernel<<<dim3(NN, BB), 256, 0, stream>>>(cur, idxbuf, invbuf, nxt,
                                                         NN, NN, 1);
    int* t = cur; cur = nxt; nxt = t;
  }

  // final layer: 2048 -> 1024, straight into d_out
  sample_kernel<<<dim3(NUM_OUT, 2), 256, 0, stream>>>(adjl, invl, idxbuf, invbuf,
                                                      NN, NUM_OUT, (unsigned)(L_HID + 1));
  nand_gather_kernel<<<dim3(NUM_OUT, BB), 256, 0, stream>>>(cur, idxbuf, invbuf, out,
                                                            NN, NUM_OUT, 1);

  (void)in_sizes; (void)n_in_args; (void)out_size; (void)ws_size;
}